// PostProcess_4011499455124
// MI455X (gfx1250) — compile-verified
//
#include <hip/hip_runtime.h>
#include <stdint.h>

typedef float f4 __attribute__((ext_vector_type(4)));

#define B_        64
#define Q_        8192
#define C_        256
#define NPB       (Q_ * C_)            // 2,097,152 elements per batch
#define NSEL      300
#define NBINS     8192                 // 13-bit key prefix
#define BIN_SHIFT 19
#define BPB       32                   // blocks per batch for streaming passes
#define TILE      (NPB / BPB)          // 65536 elements per block
#define THREADS   256
#define ITERS     (TILE / (THREADS * 4)) // 64 float4 iterations
#define CAP       2048                 // per-batch candidate capacity

// ---- order-preserving float -> uint key (monotonic increasing) ----
__device__ __forceinline__ uint32_t orderKey(float f) {
    uint32_t u = __float_as_uint(f);
    uint32_t mask = ((uint32_t)((int32_t)u >> 31)) | 0x80000000u;
    return u ^ mask;
}

// ---- CDNA5 async global -> LDS copy (ASYNCcnt path) ----
__device__ __forceinline__ void asyncLoadB128(uint32_t ldsOff, uint32_t voff, uint64_t base) {
    asm volatile("global_load_async_to_lds_b128 %0, %1, %2"
                 :
                 : "v"(ldsOff), "v"(voff), "s"(base)
                 : "memory");
}
__device__ __forceinline__ void waitAsyncLe1() {
    asm volatile("s_wait_asynccnt 0x1" ::: "memory");
}
__device__ __forceinline__ void waitAsyncLe0() {
    asm volatile("s_wait_asynccnt 0x0" ::: "memory");
}

// ---- K0: zero histogram + candidate counters ----
__global__ void __launch_bounds__(256) zero_kernel(uint32_t* __restrict__ p, int n) {
    int i = blockIdx.x * blockDim.x + threadIdx.x;
    if (i < n) p[i] = 0;
}

// ---- K1: per-batch 8192-bin histogram of key prefixes (async-to-LDS pipelined) ----
__global__ void __launch_bounds__(THREADS) hist_kernel(const float* __restrict__ logits,
                                                       uint32_t* __restrict__ hist) {
    __shared__ uint32_t lhist[NBINS];      // 32 KB
    __shared__ f4 tile[2][THREADS];        // 8 KB double buffer
    const int tid = threadIdx.x;
    const int b   = blockIdx.y;
    const int blk = blockIdx.x;

    for (int i = tid; i < NBINS; i += THREADS) lhist[i] = 0;
    __syncthreads();

    const uint64_t base = (uint64_t)(uintptr_t)(logits + (size_t)b * NPB + (size_t)blk * TILE);
    const uint32_t slot0 = (uint32_t)(uintptr_t)&tile[0][tid];
    const uint32_t slot1 = (uint32_t)(uintptr_t)&tile[1][tid];

    // prologue: stage iteration 0
    asyncLoadB128(slot0, (uint32_t)(tid * 16), base);

    for (int it = 0; it < ITERS; ++it) {
        const int buf = it & 1;
        if (it + 1 < ITERS) {
            uint32_t voff = (uint32_t)(((it + 1) * THREADS + tid) * 16);
            asyncLoadB128(buf ? slot0 : slot1, voff, base);
            waitAsyncLe1();                 // iteration 'it' complete (in-order), next in flight
        } else {
            waitAsyncLe0();
        }
        f4 v = tile[buf][tid];
        atomicAdd(&lhist[orderKey(v.x) >> BIN_SHIFT], 1u);
        atomicAdd(&lhist[orderKey(v.y) >> BIN_SHIFT], 1u);
        atomicAdd(&lhist[orderKey(v.z) >> BIN_SHIFT], 1u);
        atomicAdd(&lhist[orderKey(v.w) >> BIN_SHIFT], 1u);
    }
    __syncthreads();

    uint32_t* gh = hist + (size_t)b * NBINS;
    for (int i = tid; i < NBINS; i += THREADS) {
        uint32_t c = lhist[i];
        if (c) atomicAdd(&gh[i], c);
    }
}

// ---- K2: find threshold bin (suffix scan from the top) ----
__global__ void __launch_bounds__(256) thresh_kernel(const uint32_t* __restrict__ hist,
                                                     uint32_t* __restrict__ thresh) {
    const int b = blockIdx.x;
    const int tid = threadIdx.x;
    const uint32_t* h = hist + (size_t)b * NBINS;
    __shared__ uint32_t seg[256];

    uint32_t s = 0;
    for (int i = 0; i < 32; ++i) s += h[tid * 32 + i];
    seg[tid] = s;
    __syncthreads();

    if (tid == 0) {
        uint32_t above = 0;
        int sIdx = 0;
        {
            uint32_t acc = 0;
            for (int i = 255; i >= 0; --i) {
                uint32_t ns = acc + seg[i];
                if (ns >= NSEL) { sIdx = i; above = acc; break; }
                acc = ns;
            }
        }
        int T = sIdx * 32;
        uint32_t acc2 = above;
        for (int bin = sIdx * 32 + 31; bin >= sIdx * 32; --bin) {
            uint32_t ns = acc2 + h[bin];
            if (ns >= NSEL) { T = bin; break; }
            acc2 = ns;
        }
        thresh[b] = (uint32_t)T << BIN_SHIFT;   // lower key bound of threshold bin
    }
}

// ---- K3: compact candidates >= threshold-bin base ----
__global__ void __launch_bounds__(THREADS) compact_kernel(const float* __restrict__ logits,
                                                          const uint32_t* __restrict__ thresh,
                                                          uint32_t* __restrict__ candKey,
                                                          uint32_t* __restrict__ candIdx,
                                                          uint32_t* __restrict__ candCount) {
    const int tid = threadIdx.x;
    const int b   = blockIdx.y;
    const int blk = blockIdx.x;
    const uint32_t tk = thresh[b];
    const f4* p = (const f4*)(logits + (size_t)b * NPB + (size_t)blk * TILE);
    uint32_t* ck = candKey + (size_t)b * CAP;
    uint32_t* ci = candIdx + (size_t)b * CAP;

    for (int it = 0; it < ITERS; ++it) {
        const int e = it * THREADS + tid;
        f4 v = __builtin_nontemporal_load(&p[e]);
        const uint32_t baseIdx = (uint32_t)(blk * TILE + e * 4);
        float vals[4] = { v.x, v.y, v.z, v.w };
#pragma unroll
        for (int j = 0; j < 4; ++j) {
            uint32_t k = orderKey(vals[j]);
            if (k >= tk) {
                uint32_t pos = atomicAdd(&candCount[b], 1u);
                if (pos < CAP) { ck[pos] = k; ci[pos] = baseIdx + (uint32_t)j; }
            }
        }
    }
}

// ---- K4: bitonic sort candidates in LDS, emit top-300 scores/labels/boxes ----
__global__ void __launch_bounds__(512) select_kernel(const float* __restrict__ boxes_in,
                                                     const float* __restrict__ tsizes,
                                                     const uint32_t* __restrict__ candKey,
                                                     const uint32_t* __restrict__ candIdx,
                                                     const uint32_t* __restrict__ candCount,
                                                     float* __restrict__ outScores,
                                                     float* __restrict__ outLabels,
                                                     float* __restrict__ outBoxes) {
    __shared__ uint64_t sk[CAP];           // 16 KB: (key << 32) | ~idx
    const int b = blockIdx.x;
    const int tid = threadIdx.x;

    uint32_t cnt = candCount[b];
    if (cnt > CAP) cnt = CAP;
    const uint32_t* ck = candKey + (size_t)b * CAP;
    const uint32_t* ci = candIdx + (size_t)b * CAP;

    for (int i = tid; i < CAP; i += 512) {
        uint64_t v = 0;
        if (i < (int)cnt) v = ((uint64_t)ck[i] << 32) | (uint32_t)(~ci[i]);
        sk[i] = v;
    }
    __syncthreads();

    // bitonic sort, descending (key desc, then idx asc via ~idx)
    for (int k = 2; k <= CAP; k <<= 1) {
        for (int j = k >> 1; j > 0; j >>= 1) {
            for (int i = tid; i < CAP; i += 512) {
                int p = i ^ j;
                if (p > i) {
                    uint64_t a = sk[i], c = sk[p];
                    bool desc = ((i & k) == 0);
                    bool sw = desc ? (a < c) : (a > c);
                    if (sw) { sk[i] = c; sk[p] = a; }
                }
            }
            __syncthreads();
        }
    }

    const float img_h = tsizes[b * 2 + 0];
    const float img_w = tsizes[b * 2 + 1];

    if (tid < NSEL) {
        uint64_t v = sk[tid];
        uint32_t key = (uint32_t)(v >> 32);
        uint32_t idx = ~(uint32_t)(v & 0xFFFFFFFFu);
        uint32_t u = (key & 0x80000000u) ? (key ^ 0x80000000u) : ~key;
        float x = __uint_as_float(u);
        float score = 1.0f / (1.0f + expf(-x));
        uint32_t label = idx & (C_ - 1);
        uint32_t qb = idx >> 8;            // idx / C
        const float* pb = boxes_in + ((size_t)b * Q_ + qb) * 4;
        float cx = pb[0], cy = pb[1], w = pb[2], h = pb[3];
        float x0 = (cx - 0.5f * w) * img_w;
        float y0 = (cy - 0.5f * h) * img_h;
        float x1 = (cx + 0.5f * w) * img_w;
        float y1 = (cy + 0.5f * h) * img_h;
        outScores[b * NSEL + tid] = score;
        outLabels[b * NSEL + tid] = (float)label;
        float* ob = outBoxes + ((size_t)b * NSEL + tid) * 4;
        ob[0] = x0; ob[1] = y0; ob[2] = x1; ob[3] = y1;
    }
}

extern "C" void kernel_launch(void* const* d_in, const int* in_sizes, int n_in,
                              void* d_out, int out_size, void* d_ws, size_t ws_size,
                              hipStream_t stream) {
    const float* logits = (const float*)d_in[0];   // (64, 8192, 256) f32
    const float* boxes  = (const float*)d_in[1];   // (64, 8192, 4)   f32
    const float* tsz    = (const float*)d_in[2];   // (64, 2)         f32
    float* out = (float*)d_out;                    // scores | labels | boxes

    // workspace layout (uint32 words): hist | candCount | thresh | candKey | candIdx
    uint32_t* ws        = (uint32_t*)d_ws;
    uint32_t* hist      = ws;                                  // 64*8192
    uint32_t* candCount = ws + (size_t)B_ * NBINS;             // 64
    uint32_t* thresh    = candCount + B_;                      // 64
    uint32_t* candKey   = thresh + B_;                         // 64*2048
    uint32_t* candIdx   = candKey + (size_t)B_ * CAP;          // 64*2048

    const int zn = B_ * NBINS + B_;                            // hist + candCount (contiguous)
    zero_kernel<<<(zn + 255) / 256, 256, 0, stream>>>(ws, zn);

    dim3 gstream(BPB, B_);
    hist_kernel<<<gstream, THREADS, 0, stream>>>(logits, hist);
    thresh_kernel<<<B_, 256, 0, stream>>>(hist, thresh);
    compact_kernel<<<gstream, THREADS, 0, stream>>>(logits, thresh, candKey, candIdx, candCount);
    select_kernel<<<B_, 512, 0, stream>>>(boxes, tsz, candKey, candIdx, candCount,
                                          out, out + B_ * NSEL, out + 2 * B_ * NSEL);
}